// OVPostProcess_66322884984855
// MI455X (gfx1250) — compile-verified
//
#include <hip/hip_runtime.h>
#include <math.h>

#define Bn 32
#define Qn 900
#define Cn 80
#define Nn (Qn * Cn)          // 72000 candidates per image
#define MAX_DET 100
#define NMS_THRES 0.5f
#define SCORE_THRESH 1e-3f
#define BLOCK 1024
#define NWAVES (BLOCK / 32)

typedef float v2f __attribute__((ext_vector_type(2)));
typedef float v8f __attribute__((ext_vector_type(8)));
typedef unsigned int u32;
typedef u32 v4u __attribute__((ext_vector_type(4)));
typedef u32 v8u __attribute__((ext_vector_type(8)));

// ---- LDS layout (float offsets) -------------------------------------------
#define L_S     0                    // scores / raw logits  [72000]
#define L_BOX   (L_S + Nn)           // xyxy boxes           [900*4]
#define L_AREA  (L_BOX + Qn * 4)     // areas                [900]
#define L_VALID (L_AREA + Qn)        // per-query validity   [900]
#define L_REDV  (L_VALID + Qn)       // argmax partial vals  [32]
#define L_REDI  (L_REDV + NWAVES)    // argmax partial idxs  [32]
#define L_BC    (L_REDI + NWAVES)    // broadcast slot       [4]
#define L_TOTAL (L_BC + 4)
#define SMEM_BYTES (L_TOTAL * 4)     // 309,872 B  <= 320 KB WGP LDS

__global__ void __launch_bounds__(BLOCK)
ov_postprocess_kernel(const float* __restrict__ logits,
                      const float* __restrict__ boxes,
                      const int* __restrict__ tsize,
                      float* __restrict__ out)
{
    extern __shared__ float smem[];
    float* sS    = smem + L_S;
    float* sBox  = smem + L_BOX;
    float* sArea = smem + L_AREA;
    float* sVal  = smem + L_VALID;
    float* sRv   = smem + L_REDV;
    int*   sRi   = (int*)(smem + L_REDI);
    int*   sBc   = (int*)(smem + L_BC);

    const int b    = blockIdx.x;
    const int tid  = threadIdx.x;
    const int lane = tid & 31;
    const int wid  = tid >> 5;
    const float NEG = -__builtin_inff();

    // ---- Phase A: TDM DMA of logits[b] (900x80 f32, contiguous) -> LDS -----
    // D# group0/group1 per CDNA5 ISA ch.10.7 / 8.3-8.4. 2D tensor,
    // tile == tensor, data_size = 4B, no padding, no multicast.
    if (wid == 0) {
        u32 lds_base = (u32)(uintptr_t)((__attribute__((address_space(3))) float*)sS);
        unsigned long long ga =
            (unsigned long long)(uintptr_t)(logits + (size_t)b * Nn);
        v4u g0;
        g0[0] = 1u;                                   // count=1, user mode
        g0[1] = lds_base;                             // lds_addr
        g0[2] = (u32)(ga & 0xffffffffull);            // global_addr[31:0]
        g0[3] = (u32)((ga >> 32) & 0x01ffffffull)     // global_addr[56:32]
                | (2u << 30);                         // type = 2 ("image")
        v8u g1;
        g1[0] = (2u << 16);                           // data_size=4B, wg_mask=0
        g1[1] = ((u32)Cn) << 16;                      // tensor_dim0 = 80 (lo16)
        g1[2] = ((u32)Qn & 0xffffu) << 16;            // dim0 hi=0 | tensor_dim1=900 lo
        g1[3] = ((u32)Cn) << 16;                      // dim1 hi=0 | tile_dim0 = 80
        g1[4] = (u32)Qn & 0xffffu;                    // tile_dim1 = 900, tile_dim2 = 0
        g1[5] = (u32)Cn;                              // tensor_dim0_stride = 80
        g1[6] = 0u;                                   // stride0 hi | stride1 lo
        g1[7] = 0u;                                   // stride1 hi
        asm volatile("tensor_load_to_lds %0, %1" :: "s"(g0), "s"(g1) : "memory");
    }

    // ---- Phase B: cxcywh -> xyxy via WMMA (D = A[16x4] x B[4x16]) ----------
    // B columns 0..3 encode: x1=cx-0.5w, y1=cy-0.5h, x2=cx+0.5w, y2=cy+0.5h.
    {
        float bx, by;
        if (lane < 16) {                      // rows K=0 (cx), K=1 (cy)
            bx = (lane == 0 || lane == 2) ? 1.f : 0.f;
            by = (lane == 1 || lane == 3) ? 1.f : 0.f;
        } else {                              // rows K=2 (w), K=3 (h)
            int n = lane - 16;
            bx = (n == 0) ? -0.5f : ((n == 2) ? 0.5f : 0.f);
            by = (n == 1) ? -0.5f : ((n == 3) ? 0.5f : 0.f);
        }
        v2f Bm; Bm.x = bx; Bm.y = by;
        const int m = lane & 15;
        for (int t = wid; t < (Qn + 15) / 16; t += NWAVES) {
            const int q0 = t * 16;
            const int q  = q0 + m;
            v2f Am; Am.x = 0.f; Am.y = 0.f;
            if (q < Qn) {
                // lanes 0-15 supply (cx,cy) = K0,K1; lanes 16-31 supply (w,h) = K2,K3
                const float* p = boxes + ((size_t)b * Qn + q) * 4 + ((lane < 16) ? 0 : 2);
                Am.x = p[0]; Am.y = p[1];
            }
            v8f Cm = {};
            v8f Dm = __builtin_amdgcn_wmma_f32_16x16x4_f32(
                false, Am, false, Bm, (short)0, Cm, false, false);
            const int n = lane & 15;                  // output column (coord)
            const int rbase = (lane < 16) ? 0 : 8;    // D row = vgpr + 8*(lane>=16)
            if (n < 4) {
#pragma unroll
                for (int r = 0; r < 8; ++r) {
                    const int qq = q0 + rbase + r;
                    if (qq < Qn) sBox[qq * 4 + n] = Dm[r];
                }
            }
        }
    }
    __syncthreads();                                  // boxes visible

    // ---- areas --------------------------------------------------------------
    if (tid < Qn) {
        const float x1 = sBox[tid * 4 + 0], y1 = sBox[tid * 4 + 1];
        const float x2 = sBox[tid * 4 + 2], y2 = sBox[tid * 4 + 3];
        sArea[tid] = fmaxf(x2 - x1, 0.f) * fmaxf(y2 - y1, 0.f);
    }
    if (wid == 0) __builtin_amdgcn_s_wait_tensorcnt(0);
    __syncthreads();                                  // raw logits + areas visible

    // ---- per-query validity (finite box & all 80 logits finite) ------------
    if (tid < Qn) {
        bool v = isfinite(sBox[tid * 4 + 0]) && isfinite(sBox[tid * 4 + 1]) &&
                 isfinite(sBox[tid * 4 + 2]) && isfinite(sBox[tid * 4 + 3]);
        const float* lg = sS + tid * Cn;
        for (int c = 0; c < Cn; ++c) v = v && isfinite(lg[c]);
        sVal[tid] = v ? 1.f : 0.f;
    }
    __syncthreads();

    // ---- sigmoid + score-threshold mask, in place ---------------------------
    for (int i = tid; i < Nn; i += BLOCK) {
        const float x = sS[i];
        const float p = 1.f / (1.f + __expf(-x));
        const bool ok = (sVal[i / Cn] > 0.f) && (p > SCORE_THRESH);
        sS[i] = ok ? p : NEG;
    }
    __syncthreads();

    // ---- output pointers & scale (w,h,w,h) ----------------------------------
    float sh, sw;
    { const int h = tsize[b * 2 + 0]; const int w = tsize[b * 2 + 1];
      sh = (float)h; sw = (float)w; }
    float* out_scs  = out + (size_t)b * MAX_DET;
    float* out_lab  = out + (size_t)Bn * MAX_DET + (size_t)b * MAX_DET;
    float* out_box  = out + (size_t)2 * Bn * MAX_DET + (size_t)b * MAX_DET * 4;
    float* out_good = out + (size_t)6 * Bn * MAX_DET + (size_t)b * MAX_DET;

    // ---- greedy NMS: 100 sequential rounds ---------------------------------
    for (int it = 0; it < MAX_DET; ++it) {
        // strided scan over 72000 LDS-resident scores
        float bv = NEG; int bi = 0;
        for (int i = tid; i < Nn; i += BLOCK) {
            const float v = sS[i];
            if (v > bv) { bv = v; bi = i; }
        }
        // wave32 tree reduce, tie-break: lowest index (match jnp.argmax)
#pragma unroll
        for (int off = 16; off > 0; off >>= 1) {
            const float ov = __shfl_xor(bv, off, 32);
            const int   oi = __shfl_xor(bi, off, 32);
            if (ov > bv || (ov == bv && oi < bi)) { bv = ov; bi = oi; }
        }
        if (lane == 0) { sRv[wid] = bv; sRi[wid] = bi; }
        __syncthreads();
        if (wid == 0) {
            bv = sRv[lane]; bi = sRi[lane];
#pragma unroll
            for (int off = 16; off > 0; off >>= 1) {
                const float ov = __shfl_xor(bv, off, 32);
                const int   oi = __shfl_xor(bi, off, 32);
                if (ov > bv || (ov == bv && oi < bi)) { bv = ov; bi = oi; }
            }
            if (lane == 0) {
                const int good = isfinite(bv) ? 1 : 0;
                const int qi = bi / Cn;
                const int c  = bi - qi * Cn;
                out_scs[it] = good ? bv : 0.f;
                out_lab[it] = good ? (float)c : -1.f;
                float b0 = 0.f, b1 = 0.f, b2 = 0.f, b3 = 0.f;
                if (good) {
                    b0 = sBox[qi * 4 + 0] * sw; b1 = sBox[qi * 4 + 1] * sh;
                    b2 = sBox[qi * 4 + 2] * sw; b3 = sBox[qi * 4 + 3] * sh;
                }
                out_box[it * 4 + 0] = b0; out_box[it * 4 + 1] = b1;
                out_box[it * 4 + 2] = b2; out_box[it * 4 + 3] = b3;
                out_good[it] = good ? 1.f : 0.f;
                sBc[0] = bi; sBc[1] = c; sBc[2] = good; sBc[3] = qi;
            }
        }
        __syncthreads();
        const int imax = sBc[0], c = sBc[1], good = sBc[2], qi = sBc[3];
        if (good) {
            if (tid == 0) sS[imax] = NEG;             // idx == i suppression
            if (tid < Qn) {                           // same-class IoU suppression
                const float ax1 = sBox[qi * 4 + 0], ay1 = sBox[qi * 4 + 1];
                const float ax2 = sBox[qi * 4 + 2], ay2 = sBox[qi * 4 + 3];
                const float bx1 = sBox[tid * 4 + 0], by1 = sBox[tid * 4 + 1];
                const float bx2 = sBox[tid * 4 + 2], by2 = sBox[tid * 4 + 3];
                const float ix = fmaxf(fminf(ax2, bx2) - fmaxf(ax1, bx1), 0.f);
                const float iy = fmaxf(fminf(ay2, by2) - fmaxf(ay1, by1), 0.f);
                const float inter = ix * iy;
                const float uni = fmaxf(sArea[qi] + sArea[tid] - inter, 1e-9f);
                if (inter / uni > NMS_THRES) sS[tid * Cn + c] = NEG;
            }
        }
        __syncthreads();
    }
}

extern "C" void kernel_launch(void* const* d_in, const int* in_sizes, int n_in,
                              void* d_out, int out_size, void* d_ws, size_t ws_size,
                              hipStream_t stream) {
    (void)in_sizes; (void)n_in; (void)d_ws; (void)ws_size; (void)out_size;
    const float* logits = (const float*)d_in[0];   // [32,900,80] f32
    const float* boxes  = (const float*)d_in[1];   // [32,900,4]  f32 cxcywh
    const int*   tsize  = (const int*)d_in[2];     // [32,2] i32 (h,w)
    float* out = (float*)d_out;                    // scs|lab|boxes|goods concat

    static_assert(SMEM_BYTES <= 320 * 1024, "exceeds WGP LDS");
    hipFuncSetAttribute((const void*)ov_postprocess_kernel,
                        hipFuncAttributeMaxDynamicSharedMemorySize, SMEM_BYTES);
    ov_postprocess_kernel<<<dim3(Bn), dim3(BLOCK), SMEM_BYTES, stream>>>(
        logits, boxes, tsize, out);
}